// CFModel_91130616087237
// MI455X (gfx1250) — compile-verified
//
#include <hip/hip_runtime.h>
#include <hip/hip_bf16.h>

#define NN 50000
#define NE 800000
#define NR 16
#define DD 64

typedef float v2f __attribute__((ext_vector_type(2)));
typedef float v8f __attribute__((ext_vector_type(8)));

// ---------------------------------------------------------------------------
// out[:, 0:64] = entity_embed  (first third of the concat)
// ---------------------------------------------------------------------------
__global__ void k_copy_h0(const float* __restrict__ ent, float* __restrict__ out) {
    int i = blockIdx.x * blockDim.x + threadIdx.x;
    if (i < NN * DD) out[(size_t)(i >> 6) * 192 + (i & 63)] = ent[i];
}

// ---------------------------------------------------------------------------
// Per-layer init: h_nb = 0, segmax = -inf, segsum = 0
// ---------------------------------------------------------------------------
__global__ void k_init(float* __restrict__ hnb, float* __restrict__ m, float* __restrict__ z) {
    int i = blockIdx.x * blockDim.x + threadIdx.x;
    if (i < NN * DD) hnb[i] = 0.0f;
    if (i < NN) {
        m[i] = -__builtin_huge_valf();
        z[i] = 0.0f;
    }
}

// ---------------------------------------------------------------------------
// trans[n,r,:] = h[n,:] @ W[r]   via V_WMMA_F32_16X16X4_F32, K=64 -> 16 steps.
// Block = 128 threads = 4 waves; block b -> (row-tile b>>4, relation b&15);
// wave w computes column tile e0 = 16*w. 50000 blocks.
// ---------------------------------------------------------------------------
__global__ void k_trans_gemm(const float* __restrict__ h, const float* __restrict__ W,
                             float* __restrict__ trans) {
    int b    = blockIdx.x;
    int r    = b & 15;
    int n0   = (b >> 4) * 16;
    int lane = threadIdx.x & 31;
    int e0   = (threadIdx.x >> 5) * 16;
    int lhi  = lane >> 4;      // 0: K pair {4k,4k+1}, 1: {4k+2,4k+3}
    int llo  = lane & 15;

    const float* arow = h + (size_t)(n0 + llo) * DD + lhi * 2;
    const float* bcol = W + (size_t)r * (DD * DD) + (lhi * 2) * DD + e0 + llo;

    v8f c = {};
#pragma unroll
    for (int k = 0; k < 16; ++k) {
        v2f a = *(const v2f*)(arow + 4 * k);
        v2f bb;
        bb.x = bcol[(4 * k + 0) * DD];
        bb.y = bcol[(4 * k + 1) * DD];
        c = __builtin_amdgcn_wmma_f32_16x16x4_f32(false, a, false, bb, (short)0, c, false, false);
    }

    float* o = trans + (size_t)n0 * (NR * DD) + r * DD + e0;
#pragma unroll
    for (int j = 0; j < 8; ++j)
        o[(size_t)(j + lhi * 8) * (NR * DD) + llo] = c[j];
}

// ---------------------------------------------------------------------------
// att[e] = sum_d trans[src,r,d] * tanh(trans[dst,r,d] + rel_embed[r,d])
// wave per edge; lane handles 2 dims; butterfly reduce; lane0 writes + segmax.
// ---------------------------------------------------------------------------
__global__ void k_attn(const float* __restrict__ trans, const float* __restrict__ re,
                       const int* __restrict__ rel, const int* __restrict__ src,
                       const int* __restrict__ dst, float* __restrict__ att,
                       float* __restrict__ m) {
    int e = blockIdx.x * 8 + (threadIdx.x >> 5);
    if (e >= NE) return;
    int lane = threadIdx.x & 31;
    int r = rel[e], s = src[e], d = dst[e];

    float2 t  = *(const float2*)(trans + (size_t)s * (NR * DD) + r * DD + 2 * lane);
    float2 hh = *(const float2*)(trans + (size_t)d * (NR * DD) + r * DD + 2 * lane);
    float2 ef = *(const float2*)(re + r * DD + 2 * lane);

    float p = t.x * tanhf(hh.x + ef.x) + t.y * tanhf(hh.y + ef.y);
#pragma unroll
    for (int off = 16; off > 0; off >>= 1) p += __shfl_xor(p, off);

    if (lane == 0) {
        att[e] = p;
        // float atomic max via monotone int punning (m initialized to -inf)
        if (p >= 0.0f) atomicMax((int*)(m + d), __float_as_int(p));
        else           atomicMin((unsigned int*)(m + d), __float_as_uint(p));
    }
}

// ---------------------------------------------------------------------------
// att[e] = exp(att[e] - m[dst]);  z[dst] += att[e]
// ---------------------------------------------------------------------------
__global__ void k_expz(float* __restrict__ att, const float* __restrict__ m,
                       float* __restrict__ z, const int* __restrict__ dst) {
    int i = blockIdx.x * blockDim.x + threadIdx.x;
    if (i >= NE) return;
    int d = dst[i];
    float v = expf(att[i] - m[d]);
    att[i] = v;
    atomicAdd(z + d, v);
}

// ---------------------------------------------------------------------------
// h_nb[dst,:] += (att[e]/z[dst]) * h[src,:]   wave per edge, 2 dims/lane
// ---------------------------------------------------------------------------
__global__ void k_scatter(const float* __restrict__ att, const float* __restrict__ z,
                          const float* __restrict__ h, const int* __restrict__ src,
                          const int* __restrict__ dst, float* __restrict__ hnb) {
    int e = blockIdx.x * 8 + (threadIdx.x >> 5);
    if (e >= NE) return;
    int lane = threadIdx.x & 31;
    int s = src[e], d = dst[e];
    float a = att[e] / z[d];
    float2 hs = *(const float2*)(h + (size_t)s * DD + 2 * lane);
    atomicAdd(hnb + (size_t)d * DD + 2 * lane + 0, hs.x * a);
    atomicAdd(hnb + (size_t)d * DD + 2 * lane + 1, hs.y * a);
}

// ---------------------------------------------------------------------------
// hnext = lrelu((h+h_nb)@ra) + lrelu((h*h_nb)@rb); also write out[:,ooff:+64].
// Same WMMA tiling as k_trans_gemm; u/v built on the fly. 3125 blocks x 128.
// ---------------------------------------------------------------------------
__global__ void k_mix(const float* __restrict__ h, const float* __restrict__ hnb,
                      const float* __restrict__ ra, const float* __restrict__ rb,
                      float* __restrict__ hnext, float* __restrict__ out, int ooff) {
    int n0   = blockIdx.x * 16;
    int lane = threadIdx.x & 31;
    int e0   = (threadIdx.x >> 5) * 16;
    int lhi  = lane >> 4;
    int llo  = lane & 15;

    const float* hrow  = h   + (size_t)(n0 + llo) * DD + lhi * 2;
    const float* nrow  = hnb + (size_t)(n0 + llo) * DD + lhi * 2;
    const float* racol = ra + (lhi * 2) * DD + e0 + llo;
    const float* rbcol = rb + (lhi * 2) * DD + e0 + llo;

    v8f ca = {}, cb = {};
#pragma unroll
    for (int k = 0; k < 16; ++k) {
        v2f hv = *(const v2f*)(hrow + 4 * k);
        v2f nv = *(const v2f*)(nrow + 4 * k);
        v2f u, vv;
        u.x  = hv.x + nv.x;  u.y  = hv.y + nv.y;
        vv.x = hv.x * nv.x;  vv.y = hv.y * nv.y;
        v2f ba, bb;
        ba.x = racol[(4 * k + 0) * DD];  ba.y = racol[(4 * k + 1) * DD];
        bb.x = rbcol[(4 * k + 0) * DD];  bb.y = rbcol[(4 * k + 1) * DD];
        ca = __builtin_amdgcn_wmma_f32_16x16x4_f32(false, u,  false, ba, (short)0, ca, false, false);
        cb = __builtin_amdgcn_wmma_f32_16x16x4_f32(false, vv, false, bb, (short)0, cb, false, false);
    }

#pragma unroll
    for (int j = 0; j < 8; ++j) {
        float xa = ca[j], xb = cb[j];
        xa = xa > 0.0f ? xa : 0.01f * xa;
        xb = xb > 0.0f ? xb : 0.01f * xb;
        float v = xa + xb;
        int row = n0 + j + lhi * 8;
        int col = e0 + llo;
        hnext[(size_t)row * DD + col]        = v;
        out[(size_t)row * 192 + ooff + col]  = v;
    }
}

// ---------------------------------------------------------------------------
extern "C" void kernel_launch(void* const* d_in, const int* in_sizes, int n_in,
                              void* d_out, int out_size, void* d_ws, size_t ws_size,
                              hipStream_t stream) {
    (void)in_sizes; (void)n_in; (void)out_size; (void)ws_size;

    const int*   rel  = (const int*)d_in[1];
    const int*   src  = (const int*)d_in[2];
    const int*   dst  = (const int*)d_in[3];
    const float* ent  = (const float*)d_in[4];
    const float* re   = (const float*)d_in[5];
    const float* Wl[2]  = { (const float*)d_in[6], (const float*)d_in[7] };
    const float* ral[2] = { (const float*)d_in[8], (const float*)d_in[10] };
    const float* rbl[2] = { (const float*)d_in[9], (const float*)d_in[11] };
    float* out = (float*)d_out;

    char* w = (char*)d_ws;
    float* trans = (float*)(w);                          // 50000*16*64*4 = 204.8 MB
    float* h1    = (float*)(w + 204800000ull);           // 12.8 MB
    float* h2    = (float*)(w + 217600000ull);           // 12.8 MB
    float* hnb   = (float*)(w + 230400000ull);           // 12.8 MB
    float* att   = (float*)(w + 243200000ull);           // 3.2 MB
    float* segm  = (float*)(w + 246400000ull);           // 0.2 MB
    float* segz  = (float*)(w + 246600000ull);           // 0.2 MB

    k_copy_h0<<<(NN * DD + 255) / 256, 256, 0, stream>>>(ent, out);

    const float* hin  = ent;
    float*       hout = h1;
    for (int layer = 0; layer < 2; ++layer) {
        k_init<<<(NN * DD + 255) / 256, 256, 0, stream>>>(hnb, segm, segz);
        k_trans_gemm<<<(NN / 16) * NR, 128, 0, stream>>>(hin, Wl[layer], trans);
        k_attn<<<(NE + 7) / 8, 256, 0, stream>>>(trans, re, rel, src, dst, att, segm);
        k_expz<<<(NE + 255) / 256, 256, 0, stream>>>(att, segm, segz, dst);
        k_scatter<<<(NE + 7) / 8, 256, 0, stream>>>(att, segz, hin, src, dst, hnb);
        k_mix<<<NN / 16, 128, 0, stream>>>(hin, hnb, ral[layer], rbl[layer],
                                           hout, out, DD + layer * DD);
        hin  = hout;
        hout = h2;
    }
}